// SAGE_59811714564516
// MI455X (gfx1250) — compile-verified
//
#include <hip/hip_runtime.h>
#include <math.h>

typedef __attribute__((ext_vector_type(2))) float v2f;
typedef __attribute__((ext_vector_type(8))) float v8f;

#define N0c 585728
#define N1c 22528
#define N2c 2048
#define E1c 563200
#define E2c 20480
#define INC 602
#define HIDc 256
#define OUTC 41

// ---------------- zero ----------------
__global__ void zero_f32(float* __restrict__ p, long n) {
    long i = (long)blockIdx.x * blockDim.x + threadIdx.x;
    long stride = (long)gridDim.x * blockDim.x;
    for (; i < n; i += stride) p[i] = 0.0f;
}

// -------- reciprocal of clamped counts: cnt[i] = 1/max(cnt[i],1) --------
__global__ void inv_cnt(float* __restrict__ cnt, int n) {
    int i = blockIdx.x * blockDim.x + threadIdx.x;
    if (i < n) cnt[i] = 1.0f / fmaxf(cnt[i], 1.0f);
}

// ---------------- layer-1 scatter (sum numerator + counts) ----------------
__global__ void scatter1(const float* __restrict__ x, const int* __restrict__ src,
                         const int* __restrict__ dst, float* __restrict__ agg,
                         float* __restrict__ cnt) {
    int e = blockIdx.x;
    int s = src[e];
    int d = dst[e];
    const float* xr = x + (long)s * INC;
    float* ar = agg + (long)d * INC;
    for (int c = threadIdx.x; c < INC; c += blockDim.x)
        atomicAdd(&ar[c], xr[c]);
    if (threadIdx.x == 0) atomicAdd(&cnt[d], 1.0f);
}

// ---------------- layer-2 scatter ----------------
__global__ void scatter2(const float* __restrict__ h, const int* __restrict__ src,
                         const int* __restrict__ dst, float* __restrict__ agg,
                         float* __restrict__ cnt) {
    int e = blockIdx.x;
    int s = src[e];
    int d = dst[e];
    atomicAdd(&agg[(long)d * HIDc + threadIdx.x], h[(long)s * HIDc + threadIdx.x]);
    if (threadIdx.x == 0) atomicAdd(&cnt[d], 1.0f);
}

// ---------------- layer 1: h = relu((agg*inv)@Wl + bias + xdst@Wr) ------
// One wave per 16x16 output tile, 8 waves/block along M, all waves share the
// N-tile so both weight panels are staged (transposed) in LDS once per block.
// A frag (16x4 f32): lane<16 -> M=lane,   K=k0+{0,1};
//                    lane>=16 -> M=lane-16, K=k0+{2,3}.
// B frag (4x16 f32): lane<16 -> N=lane,   K=k0+{0,1};
//                    lane>=16 -> N=lane-16, K=k0+{2,3}.
// C/D (16x16 f32): vgpr i, lane<16 -> (M=i, N=lane); lane>=16 -> (M=i+8, N=lane-16)
#define B1STR 606   // even (b64-aligned), spreads LDS bank pairs
__global__ __launch_bounds__(256)
void sage_gemm1(const float* __restrict__ agg,    // [N1, 602] (sums)
                const float* __restrict__ x,      // [N0, 602] (rows < N1 used)
                const float* __restrict__ Wl,     // [602, 256]
                const float* __restrict__ Wr,     // [602, 256]
                const float* __restrict__ bias,   // [256]
                const float* __restrict__ cntInv, // [N1] 1/max(deg,1)
                float* __restrict__ h)            // [N1, 256]
{
    __shared__ float ldsWl[16 * B1STR];
    __shared__ float ldsWr[16 * B1STR];

    const int tid   = threadIdx.x;
    const int nBase = blockIdx.x * 16;

    // stage transposed panels: lds[c*B1STR + k] = W[k*256 + nBase + c]
    // (idx = k*16 + c so consecutive threads read consecutive floats)
    for (int idx = tid; idx < INC * 16; idx += 256) {
        int k = idx >> 4;
        int c = idx & 15;
        ldsWl[c * B1STR + k] = Wl[(long)k * HIDc + nBase + c];
        ldsWr[c * B1STR + k] = Wr[(long)k * HIDc + nBase + c];
    }
    __syncthreads();

    const int lane  = tid & 31;
    const int wave  = tid >> 5;
    const int mBase = (blockIdx.y * 8 + wave) * 16;
    const int mr    = lane & 15;
    const int kOff  = (lane >> 4) * 2;
    const bool lo   = (kOff == 0);

    const float* aPtr  = agg + (long)(mBase + mr) * INC + kOff;
    const float* xPtr  = x   + (long)(mBase + mr) * INC + kOff;
    const float* aTail = agg + (long)(mBase + mr) * INC + 600;
    const float* xTail = x   + (long)(mBase + mr) * INC + 600;
    const float* bl    = ldsWl + mr * B1STR + kOff;
    const float* br    = ldsWr + mr * B1STR + kOff;
    const float  scA   = cntInv[mBase + mr];

    v8f acc = {};
    // ---- pass 1: (agg * scA) @ Wl ----
#pragma unroll 2
    for (int k0 = 0; k0 < 600; k0 += 4) {
        v2f a = *(const v2f*)(aPtr + k0) * scA;
        v2f b = *(const v2f*)(bl + k0);
        acc = __builtin_amdgcn_wmma_f32_16x16x4_f32(
            false, a, false, b, (short)0, acc, false, false);
    }
    {   // tail K = 600,601 (only lane<16 half contributes; branch-free masks)
        float t0 = aTail[0], t1 = aTail[1];
        float w0 = ldsWl[mr * B1STR + 600], w1 = ldsWl[mr * B1STR + 601];
        v2f a, b;
        a.x = lo ? t0 * scA : 0.0f;  a.y = lo ? t1 * scA : 0.0f;
        b.x = lo ? w0 : 0.0f;        b.y = lo ? w1 : 0.0f;
        acc = __builtin_amdgcn_wmma_f32_16x16x4_f32(
            false, a, false, b, (short)0, acc, false, false);
    }
    // ---- pass 2: x_dst @ Wr ----
#pragma unroll 2
    for (int k0 = 0; k0 < 600; k0 += 4) {
        v2f a = *(const v2f*)(xPtr + k0);
        v2f b = *(const v2f*)(br + k0);
        acc = __builtin_amdgcn_wmma_f32_16x16x4_f32(
            false, a, false, b, (short)0, acc, false, false);
    }
    {
        float t0 = xTail[0], t1 = xTail[1];
        float w0 = ldsWr[mr * B1STR + 600], w1 = ldsWr[mr * B1STR + 601];
        v2f a, b;
        a.x = lo ? t0 : 0.0f;  a.y = lo ? t1 : 0.0f;
        b.x = lo ? w0 : 0.0f;  b.y = lo ? w1 : 0.0f;
        acc = __builtin_amdgcn_wmma_f32_16x16x4_f32(
            false, a, false, b, (short)0, acc, false, false);
    }

    const float bn   = bias[nBase + mr];
    const int   mOff = (lane >> 4) * 8;
    float* hOut = h + (long)(mBase + mOff) * HIDc + nBase + mr;
#pragma unroll
    for (int i = 0; i < 8; ++i)
        hOut[(long)i * HIDc] = fmaxf(acc[i] + bn, 0.0f);
}

// ---------------- layer 2: logits = (agg2*inv)@Wl2 + bias + hdst@Wr2 ----
// Same scheme; col>=41 padding applied once at LDS staging -> guard-free loop.
#define B2STR 258
__global__ __launch_bounds__(256)
void sage_gemm2(const float* __restrict__ agg2,    // [N2, 256] (sums)
                const float* __restrict__ h,       // [N1, 256] (rows < N2 used)
                const float* __restrict__ Wl,      // [256, 41]
                const float* __restrict__ Wr,      // [256, 41]
                const float* __restrict__ bias,    // [41]
                const float* __restrict__ cntInv,  // [N2]
                float* __restrict__ logits)        // [N2, 41]
{
    __shared__ float ldsWl[16 * B2STR];
    __shared__ float ldsWr[16 * B2STR];

    const int tid   = threadIdx.x;
    const int nBase = blockIdx.x * 16;

    for (int idx = tid; idx < HIDc * 16; idx += 256) {
        int k = idx >> 4;
        int c = idx & 15;
        int col = nBase + c;
        bool ok = (col < OUTC);
        ldsWl[c * B2STR + k] = ok ? Wl[(long)k * OUTC + col] : 0.0f;
        ldsWr[c * B2STR + k] = ok ? Wr[(long)k * OUTC + col] : 0.0f;
    }
    __syncthreads();

    const int lane  = tid & 31;
    const int wave  = tid >> 5;
    const int mBase = (blockIdx.y * 8 + wave) * 16;
    const int mr    = lane & 15;
    const int kOff  = (lane >> 4) * 2;

    const float* aPtr = agg2 + (long)(mBase + mr) * HIDc + kOff;
    const float* hPtr = h    + (long)(mBase + mr) * HIDc + kOff;
    const float* bl   = ldsWl + mr * B2STR + kOff;
    const float* br   = ldsWr + mr * B2STR + kOff;
    const float  scA  = cntInv[mBase + mr];

    v8f acc = {};
#pragma unroll 2
    for (int k0 = 0; k0 < HIDc; k0 += 4) {
        v2f a = *(const v2f*)(aPtr + k0) * scA;
        v2f b = *(const v2f*)(bl + k0);
        acc = __builtin_amdgcn_wmma_f32_16x16x4_f32(
            false, a, false, b, (short)0, acc, false, false);
    }
#pragma unroll 2
    for (int k0 = 0; k0 < HIDc; k0 += 4) {
        v2f a = *(const v2f*)(hPtr + k0);
        v2f b = *(const v2f*)(br + k0);
        acc = __builtin_amdgcn_wmma_f32_16x16x4_f32(
            false, a, false, b, (short)0, acc, false, false);
    }

    const int  col   = nBase + mr;
    const bool colOk = (col < OUTC);
    const float bn   = colOk ? bias[col] : 0.0f;
    const int  mOff  = (lane >> 4) * 8;
#pragma unroll
    for (int i = 0; i < 8; ++i) {
        int m = mBase + mOff + i;
        if (colOk) logits[(long)m * OUTC + col] = acc[i] + bn;
    }
}

// ---------------- log_softmax over 41 logits, one wave per row ----------
__global__ void logsoftmax41(const float* __restrict__ logits, float* __restrict__ out) {
    const int row  = blockIdx.x;
    const int lane = threadIdx.x;   // blockDim.x == 32 (wave32)
    const float* r = logits + (long)row * OUTC;

    float v0 = (lane      < OUTC) ? r[lane]      : -INFINITY;
    float v1 = (lane + 32 < OUTC) ? r[lane + 32] : -INFINITY;

    float m = fmaxf(v0, v1);
    for (int off = 16; off > 0; off >>= 1)
        m = fmaxf(m, __shfl_xor(m, off, 32));

    float s = ((lane      < OUTC) ? expf(v0 - m) : 0.0f) +
              ((lane + 32 < OUTC) ? expf(v1 - m) : 0.0f);
    for (int off = 16; off > 0; off >>= 1)
        s += __shfl_xor(s, off, 32);

    const float lse = m + logf(s);
    float* o = out + (long)row * OUTC;
    if (lane      < OUTC) o[lane]      = v0 - lse;
    if (lane + 32 < OUTC) o[lane + 32] = v1 - lse;
}

extern "C" void kernel_launch(void* const* d_in, const int* in_sizes, int n_in,
                              void* d_out, int out_size, void* d_ws, size_t ws_size,
                              hipStream_t stream) {
    const float* x    = (const float*)d_in[0];
    const int*   ei1  = (const int*)d_in[1];   // [2, E1] flat: row0 src, row1 dst
    const int*   ei2  = (const int*)d_in[2];
    const float* Wl1  = (const float*)d_in[3];
    const float* bl1  = (const float*)d_in[4];
    const float* Wr1  = (const float*)d_in[5];
    const float* Wl2  = (const float*)d_in[6];
    const float* bl2  = (const float*)d_in[7];
    const float* Wr2  = (const float*)d_in[8];
    float* out = (float*)d_out;

    // workspace layout (floats)
    float* agg1   = (float*)d_ws;                       // N1*602
    float* cnt1   = agg1 + (size_t)N1c * INC;           // N1
    float* hbuf   = cnt1 + N1c;                         // N1*256
    float* agg2   = hbuf + (size_t)N1c * HIDc;          // N2*256
    float* cnt2   = agg2 + (size_t)N2c * HIDc;          // N2
    float* logits = cnt2 + N2c;                         // N2*41

    const int* src1 = ei1;
    const int* dst1 = ei1 + E1c;
    const int* src2 = ei2;
    const int* dst2 = ei2 + E2c;

    // 1) zero accumulators (agg1+cnt1 contiguous, agg2+cnt2 contiguous)
    zero_f32<<<2048, 256, 0, stream>>>(agg1, (long)N1c * INC + N1c);
    zero_f32<<<512, 256, 0, stream>>>(agg2, (long)N2c * HIDc + N2c);

    // 2) layer-1 neighbor scatter, then invert counts (mean folded into GEMM)
    scatter1<<<E1c, 128, 0, stream>>>(x, src1, dst1, agg1, cnt1);
    inv_cnt<<<(N1c + 255) / 256, 256, 0, stream>>>(cnt1, N1c);

    // 3) layer-1 fused dual GEMM + bias + relu  (WMMA f32 16x16x4, LDS B-panels)
    sage_gemm1<<<dim3(HIDc / 16, N1c / 128), 256, 0, stream>>>(
        agg1, x, Wl1, Wr1, bl1, cnt1, hbuf);

    // 4) layer-2 scatter + count inversion
    scatter2<<<E2c, HIDc, 0, stream>>>(hbuf, src2, dst2, agg2, cnt2);
    inv_cnt<<<(N2c + 255) / 256, 256, 0, stream>>>(cnt2, N2c);

    // 5) layer-2 fused dual GEMM + bias  (N padded in LDS to 3 tiles of 16)
    sage_gemm2<<<dim3(3, N2c / 128), 256, 0, stream>>>(
        agg2, hbuf, Wl2, Wr2, bl2, cnt2, logits);

    // 6) row-wise log_softmax
    logsoftmax41<<<N2c, 32, 0, stream>>>(logits, out);
}